// beta_net_14843406975133
// MI455X (gfx1250) — compile-verified
//
#include <hip/hip_runtime.h>
#include <hip/hip_bf16.h>
#include <cstdint>

// Problem constants (fixed by setup_inputs)
#define T_TOT   128
#define B_TOT   1024
#define DDIM    512
#define NPAST   16
#define POSV    128
#define SEG     (POSV - NPAST)     // 112
#define NCHUNK  8
#define TCHUNK  (SEG / NCHUNK)     // 14
#define BG      16                 // batch rows per workgroup
#define LSTR    516                // padded LDS row stride (floats): bank-conflict-free WMMA A reads

typedef float v8f __attribute__((ext_vector_type(8)));
typedef float v2f __attribute__((ext_vector_type(2)));
typedef int   gccv4i __attribute__((vector_size(16)));   // matches builtin prototype V4i

// ---------------- CDNA5 async global->LDS (double-buffered tile DMA) ----------------
#if __has_builtin(__builtin_amdgcn_global_load_async_to_lds_b128) && __has_builtin(__builtin_amdgcn_s_wait_asynccnt)
#define USE_ASYNC 1
#else
#define USE_ASYNC 0
#endif

__device__ __forceinline__ void cp16(const float* g, float* l) {
#if USE_ASYNC
  __builtin_amdgcn_global_load_async_to_lds_b128(
      (__attribute__((address_space(1))) gccv4i*)(uintptr_t)g,
      (__attribute__((address_space(3))) gccv4i*)(uintptr_t)l,
      0, 0);
#else
  *(float4*)l = *(const float4*)g;
#endif
}
__device__ __forceinline__ void async_wait0() {
#if USE_ASYNC
  __builtin_amdgcn_s_wait_asynccnt(0);
#endif
}

#if __has_builtin(__builtin_amdgcn_wmma_f32_16x16x4_f32)
#define HAVE_WMMA_F32 1
#else
#define HAVE_WMMA_F32 0
#endif

// ---------------- Kernel 0: bias[b] = h.w5 + w5b + (1/16) sum_past(h_list.w6) + w6b ----------------
__global__ __launch_bounds__(128) void k_bias(
    const float* __restrict__ hl, const float* __restrict__ h,
    const float* __restrict__ w5w, const float* __restrict__ w5b,
    const float* __restrict__ w6w, const float* __restrict__ w6b,
    float* __restrict__ biasv)
{
  __shared__ float red[4];
  const int tid = threadIdx.x, lane = tid & 31, wave = tid >> 5;
  const int b = blockIdx.x;

  float p5 = 0.f, p6 = 0.f;
  for (int d = tid; d < DDIM; d += 128)
    p5 = fmaf(h[(size_t)b * DDIM + d], w5w[d], p5);
  for (int t = 0; t < NPAST; ++t) {
    const float* row = hl + ((size_t)t * B_TOT + b) * DDIM;
    for (int d = tid; d < DDIM; d += 128)
      p6 = fmaf(row[d], w6w[d], p6);
  }
  float p = p5 + p6 * (1.f / (float)NPAST);
#pragma unroll
  for (int off = 16; off >= 1; off >>= 1) p += __shfl_down(p, off);
  if (lane == 0) red[wave] = p;
  __syncthreads();
  if (tid == 0)
    biasv[b] = red[0] + red[1] + red[2] + red[3] + w5b[0] + w6b[0];
}

// ---------------- Kernel 1: single-pass streaming over seg with online num/den ----------------
// Grid: 512 blocks = 64 batch-groups x 8 t-chunks. Block: 256 threads (8 waves).
// Per t: stage 16x512 f32 tile in LDS (async DMA, double-buffered), 16 dots via
// v_wmma_f32_16x16x4_f32 (8 waves x 64-wide K slices), then weighted accumulate.
__global__ __launch_bounds__(256) void k_chunk(
    const float* __restrict__ hl, const float* __restrict__ w4w,
    const float* __restrict__ w4b, const float* __restrict__ biasv,
    float* __restrict__ accP, float* __restrict__ denP)
{
  __shared__ float tile[2 * BG * LSTR];   // 66048 B double-buffered tile
  __shared__ float w4s[DDIM];
  __shared__ float dotp[8][BG];

  const int tid  = threadIdx.x;
  const int wave = tid >> 5, lane = tid & 31;
  const int bsub = tid >> 4, dlane = tid & 15;
  const int hi = lane >> 4, m = lane & 15;
  const int bg    = blockIdx.x >> 3;     // 0..63
  const int chunk = blockIdx.x & 7;      // 0..7
  const int b0 = bg * BG;
  const int t0 = NPAST + chunk * TCHUNK;

  // Stage w4 into LDS
  w4s[tid]       = w4w[tid];
  w4s[tid + 256] = w4w[tid + 256];

  // Issue first tile (row = bsub, 8 x b128 per thread)
  {
    const float* g = hl + ((size_t)t0 * B_TOT + b0 + bsub) * DDIM + 4 * dlane;
    float* l = tile + bsub * LSTR + 4 * dlane;
#pragma unroll
    for (int k = 0; k < 8; ++k) cp16(g + 64 * k, l + 64 * k);
  }
  async_wait0();
  __syncthreads();

  const float w4bias = w4b[0];
  const float biasr  = biasv[b0 + bsub];

#if HAVE_WMMA_F32
  // B operand: w4 K-slice broadcast into all 16 columns.
  // B 4x16 f32 layout: lane l VGPR v holds B[K = v + 2*(l>>4)][N = l&15].
  v2f breg[16];
#pragma unroll
  for (int j = 0; j < 16; ++j) {
    int kb = 64 * wave + 4 * j + 2 * hi;
    breg[j].x = w4s[kb];
    breg[j].y = w4s[kb + 1];
  }
#endif

  float4 acc[8];
#pragma unroll
  for (int k = 0; k < 8; ++k) acc[k] = make_float4(0.f, 0.f, 0.f, 0.f);
  float den = 0.f;

  for (int it = 0; it < TCHUNK; ++it) {
    const float* tp = tile + (it & 1) * (BG * LSTR);

    // Prefetch next tile into the other buffer (overlaps with compute below)
    if (it + 1 < TCHUNK) {
      const int t = t0 + it + 1;
      const float* g = hl + ((size_t)t * B_TOT + b0 + bsub) * DDIM + 4 * dlane;
      float* l = tile + ((it + 1) & 1) * (BG * LSTR) + bsub * LSTR + 4 * dlane;
#pragma unroll
      for (int k = 0; k < 8; ++k) cp16(g + 64 * k, l + 64 * k);
    }

    // ---- Phase A: dots[bb] = seg[bb,:] . w4, wave covers K in [64*wave, 64*wave+64) ----
#if HAVE_WMMA_F32
    {
      v8f C = {};
      const int rowoff = m * LSTR + 64 * wave + 2 * hi;  // A: lane m=l&15, K=2*hi+{0,1}
#pragma unroll
      for (int j = 0; j < 16; ++j) {
        v2f a = *(const v2f*)(tp + rowoff + 4 * j);
        C = __builtin_amdgcn_wmma_f32_16x16x4_f32(
                false, a, false, breg[j], (short)0, C, false, false);
      }
      // D layout: VGPR j -> M=j (lanes 0-15), M=8+j (lanes 16-31); take column N=0.
      if (lane == 0) {
#pragma unroll
        for (int j = 0; j < 8; ++j) dotp[wave][j] = C[j];
      } else if (lane == 16) {
#pragma unroll
        for (int j = 0; j < 8; ++j) dotp[wave][8 + j] = C[j];
      }
    }
#else
    {
      float p = 0.f;
      const int kb = 64 * wave + 32 * hi;
      const float* rp = tp + m * LSTR + kb;
#pragma unroll
      for (int k = 0; k < 32; ++k) p = fmaf(rp[k], w4s[kb + k], p);
      p += __shfl_down(p, 16);
      if (lane < 16) dotp[wave][m] = p;
    }
#endif
    __syncthreads();

    float dot = 0.f;
#pragma unroll
    for (int w = 0; w < 8; ++w) dot += dotp[w][bsub];
    const float s = __expf(tanhf(dot + w4bias + biasr));
    den += s;

    // ---- Phase B: acc[bsub, d] += s * seg[bsub, d] (thread owns d = 4*dlane + 64k + j) ----
    const float* rp = tp + bsub * LSTR + 4 * dlane;
#pragma unroll
    for (int k = 0; k < 8; ++k) {
      float4 v = *(const float4*)(rp + 64 * k);
      acc[k].x = fmaf(s, v.x, acc[k].x);
      acc[k].y = fmaf(s, v.y, acc[k].y);
      acc[k].z = fmaf(s, v.z, acc[k].z);
      acc[k].w = fmaf(s, v.w, acc[k].w);
    }

    if (it + 1 < TCHUNK) async_wait0();
    __syncthreads();   // dotp reuse + buffer swap safety
  }

  // Write per-chunk partials
  {
    const size_t base = ((size_t)chunk * B_TOT + b0 + bsub) * DDIM + 4 * dlane;
#pragma unroll
    for (int k = 0; k < 8; ++k) *(float4*)(accP + base + 64 * k) = acc[k];
    if (dlane == 0) denP[chunk * B_TOT + b0 + bsub] = den;
  }
}

// ---------------- Kernel 2: combine chunk partials (deterministic fixed-order sums) ----------------
__global__ __launch_bounds__(256) void k_combine(
    const float* __restrict__ accP, const float* __restrict__ denP,
    float* __restrict__ out)
{
  const int idx = blockIdx.x * 256 + threadIdx.x;   // < B*D
  const int b = idx >> 9;
  const size_t BD = (size_t)B_TOT * DDIM;
  float den = 0.f, num = 0.f;
#pragma unroll
  for (int c = 0; c < NCHUNK; ++c) {
    den += denP[(size_t)c * B_TOT + b];
    num += accP[(size_t)c * BD + idx];
  }
  out[idx] = num / den;
}

extern "C" void kernel_launch(void* const* d_in, const int* in_sizes, int n_in,
                              void* d_out, int out_size, void* d_ws, size_t ws_size,
                              hipStream_t stream)
{
  const float* hl  = (const float*)d_in[0];  // (128,1024,512)
  const float* h   = (const float*)d_in[1];  // (1024,512)
  const float* w4w = (const float*)d_in[2];  // (512,1)
  const float* w4b = (const float*)d_in[3];  // (1,)
  const float* w5w = (const float*)d_in[4];
  const float* w5b = (const float*)d_in[5];
  const float* w6w = (const float*)d_in[6];
  const float* w6b = (const float*)d_in[7];
  float* out = (float*)d_out;

  // Workspace layout (floats): bias[1024] | denP[8*1024] | accP[8*1024*512]  (~16.8 MB)
  float* ws    = (float*)d_ws;
  float* biasv = ws;
  float* denP  = ws + B_TOT;
  float* accP  = ws + B_TOT + NCHUNK * B_TOT;

  k_bias<<<B_TOT, 128, 0, stream>>>(hl, h, w5w, w5b, w6w, w6b, biasv);
  k_chunk<<<NCHUNK * (B_TOT / BG), 256, 0, stream>>>(hl, w4w, w4b, biasv, accP, denP);
  k_combine<<<(B_TOT * DDIM) / 256, 256, 0, stream>>>(accP, denP, out);
}